// CROSS_SS2D_32890859553446
// MI455X (gfx1250) — compile-verified
//
#include <hip/hip_runtime.h>
#include <math.h>

// Problem constants (must match reference)
#define B_  2
#define H_  32
#define W_  32
#define DM_ 96
#define DI_ 192
#define N_  16
#define R_  6
#define K_  4
#define L_  1024
#define TL_ 64              // scan tile length (steps staged in LDS per buffer)

typedef __attribute__((ext_vector_type(2))) float v2f;
typedef __attribute__((ext_vector_type(8))) float v8f;

static __device__ __forceinline__ v8f wmma4(v2f a, v2f b, v8f c) {
  // D = A(16x4,f32) * B(4x16,f32) + C(16x16,f32)  -> v_wmma_f32_16x16x4_f32
  return __builtin_amdgcn_wmma_f32_16x16x4_f32(false, a, false, b, (short)0, c,
                                               false, false);
}

static __device__ __forceinline__ float silu_(float x) {
  return x / (1.0f + expf(-x));
}

// direction permutation: natural index p for direction-kd step l
static __device__ __forceinline__ int perm_k(int kd, int l) {
  if (kd == 0) return l;
  if (kd == 1) return (l & 31) * 32 + (l >> 5);
  if (kd == 2) return (L_ - 1) - l;
  const int l1 = (L_ - 1) - l;
  return (l1 & 31) * 32 + (l1 >> 5);
}

// Async global -> LDS copy of one dword (gfx1250, ASYNCcnt-tracked).
static __device__ __forceinline__ void async_ld_b32(unsigned lds_byte,
                                                    const float* gaddr) {
  asm volatile("global_load_async_to_lds_b32 %0, %1, off"
               :: "v"(lds_byte), "v"(gaddr)
               : "memory");
}
static __device__ __forceinline__ void wait_asynccnt0() {
  asm volatile("s_wait_asynccnt 0x0" ::: "memory");
}

// ---------------------------------------------------------------------------
// Kernel 1: in-projection GEMM, one branch per launch.
//   xz[i,o] = sum_c x[i,c] * w[o,c],  i in [0,2048), o in [0,384)
//   o <  192 -> xc_raw[b, o, p]     ((B,DI,L) layout, conv input)
//   o >= 192 -> z[b, p, o-192]      ((B,L,DI) layout)
// grid (128 Mtiles, 24 Ntiles), block 32 (one wave per 16x16 tile)
// ---------------------------------------------------------------------------
__global__ __launch_bounds__(32) void k_inproj(const float* __restrict__ x,
                                               const float* __restrict__ w,
                                               float* __restrict__ xc_raw,
                                               float* __restrict__ z) {
  const int lane = threadIdx.x & 31;
  const int hf = lane >> 4;       // 0: lanes 0-15, 1: lanes 16-31
  const int lr = lane & 15;
  const int m0 = blockIdx.x * 16; // row i
  const int n0 = blockIdx.y * 16; // col o
  v8f c = {};
  const float* arow = x + (size_t)(m0 + lr) * DM_;
  const float* brow = w + (size_t)(n0 + lr) * DM_;  // B[k][n] = w[n][k]
#pragma unroll 4
  for (int k0 = 0; k0 < DM_; k0 += 4) {
    v2f a, b;
    const int kb = k0 + 2 * hf;
    a.x = arow[kb];     a.y = arow[kb + 1];
    b.x = brow[kb];     b.y = brow[kb + 1];
    c = wmma4(a, b, c);
  }
  const int o = n0 + lr;
#pragma unroll
  for (int v = 0; v < 8; ++v) {
    const int i = m0 + v + 8 * hf;          // global row
    const int bi = i >> 10;                 // batch
    const int p = i & (L_ - 1);             // spatial
    const float val = c[v];
    if (o < DI_)
      xc_raw[((size_t)(bi * DI_ + o) << 10) + p] = val;
    else
      z[(((size_t)bi << 10) + p) * DI_ + (o - DI_)] = val;
  }
}

// ---------------------------------------------------------------------------
// Kernel 2: depthwise 3x3 conv (pad 1) + bias + SiLU, one branch per launch.
// in/out layout (B, DI, L).  grid = B*DI*L/256, block 256.
// ---------------------------------------------------------------------------
__global__ __launch_bounds__(256) void k_conv_silu(const float* __restrict__ in,
                                                   const float* __restrict__ w3,
                                                   const float* __restrict__ bias,
                                                   float* __restrict__ out) {
  const int idx = blockIdx.x * 256 + threadIdx.x;   // < B*DI*L
  const int p = idx & (L_ - 1);
  const int bd = idx >> 10;         // b*DI + d
  const int d = bd % DI_;
  const int h = p >> 5, wc = p & 31;
  float acc = bias[d];
  const float* base = in + ((size_t)bd << 10);
  const float* wf = w3 + d * 9;
#pragma unroll
  for (int dy = 0; dy < 3; ++dy) {
    const int yy = h + dy - 1;
    if (yy < 0 || yy >= H_) continue;
#pragma unroll
    for (int dx = 0; dx < 3; ++dx) {
      const int xx = wc + dx - 1;
      if (xx < 0 || xx >= W_) continue;
      acc += wf[dy * 3 + dx] * base[yy * W_ + xx];
    }
  }
  out[(size_t)idx] = silu_(acc);
}

// ---------------------------------------------------------------------------
// Kernel 3: cross-modal 1x1 fuse GEMM.
//   x_fuse[b,o,p] = sum_{c<576} fuse_w[o,c] * concat(xTC,xVC,xVG)[b,c,p]
// grid (12 Mtiles(o), 64 Ntiles(p), B), block 32.
// ---------------------------------------------------------------------------
__global__ __launch_bounds__(32) void k_fuse(const float* __restrict__ fw,
                                             const float* __restrict__ xa0,
                                             const float* __restrict__ xa1,
                                             const float* __restrict__ xa2,
                                             float* __restrict__ xf) {
  const int lane = threadIdx.x & 31;
  const int hf = lane >> 4, lr = lane & 15;
  const int m0 = blockIdx.x * 16;   // o
  const int n0 = blockIdx.y * 16;   // p
  const int bi = blockIdx.z;
  v8f c = {};
  const float* arow = fw + (size_t)(m0 + lr) * (3 * DI_);
  for (int k0 = 0; k0 < 3 * DI_; k0 += 4) {
    const int kb = k0 + 2 * hf;         // absolute channel of b.x
    const int t = kb / DI_;             // branch (group of 4 never crosses)
    const int ch = kb - t * DI_;
    const float* xa = (t == 0) ? xa0 : ((t == 1) ? xa1 : xa2);
    v2f a, b;
    a.x = arow[kb];     a.y = arow[kb + 1];
    const size_t bb = ((size_t)(bi * DI_ + ch) << 10) + (n0 + lr);
    b.x = xa[bb];
    b.y = xa[bb + L_];                  // next channel, same p
    c = wmma4(a, b, c);
  }
#pragma unroll
  for (int v = 0; v < 8; ++v) {
    const int o = m0 + v + 8 * hf;
    xf[((size_t)(bi * DI_ + o) << 10) + (n0 + lr)] = c[v];
  }
}

// ---------------------------------------------------------------------------
// Kernel 4: x_proj GEMM, natural l-order.
//   Ynat[b,k,c,p] = sum_d x_proj_w[k,c,d] * x_fuse[b,d,p],  c in [0,38)
// grid (3 Mtiles(c, padded to 48), 64 Ntiles(p), B*K), block 32.
// ---------------------------------------------------------------------------
__global__ __launch_bounds__(32) void k_xdbl(const float* __restrict__ xpw,
                                             const float* __restrict__ xf,
                                             float* __restrict__ ynat) {
  const int lane = threadIdx.x & 31;
  const int hf = lane >> 4, lr = lane & 15;
  const int m0 = blockIdx.x * 16;   // c
  const int n0 = blockIdx.y * 16;   // p
  const int bk = blockIdx.z;        // b*K + k
  const int kd = bk & 3;
  const int bi = bk >> 2;
  const int C_ = R_ + 2 * N_;       // 38
  v8f c = {};
  const int arow = m0 + lr;
  const float* ap = xpw + ((size_t)kd * C_ + arow) * DI_;
  const bool avalid = (arow < C_);
  for (int k0 = 0; k0 < DI_; k0 += 4) {
    const int kb = k0 + 2 * hf;
    v2f a, b;
    a.x = avalid ? ap[kb] : 0.0f;
    a.y = avalid ? ap[kb + 1] : 0.0f;
    const size_t bb = ((size_t)(bi * DI_ + kb) << 10) + (n0 + lr);
    b.x = xf[bb];
    b.y = xf[bb + L_];
    c = wmma4(a, b, c);
  }
#pragma unroll
  for (int v = 0; v < 8; ++v) {
    const int crow = m0 + v + 8 * hf;
    if (crow < C_)
      ynat[(((size_t)bk * C_ + crow) << 10) + (n0 + lr)] = c[v];
  }
}

// ---------------------------------------------------------------------------
// Kernel 5: dt projection + softplus (K-dim only 6 -> scalar).
//   delta[b,k,d,p] = softplus(sum_r dt_w[k,d,r]*Ynat[b,k,r,p] + dt_b[k,d])
// grid = B*K*DI*L/256, block 256.
// ---------------------------------------------------------------------------
__global__ __launch_bounds__(256) void k_delta(const float* __restrict__ ynat,
                                               const float* __restrict__ dt_w,
                                               const float* __restrict__ dt_b,
                                               float* __restrict__ delta) {
  const int idx = blockIdx.x * 256 + threadIdx.x;  // < B*K*DI*L
  const int p = idx & (L_ - 1);
  const int rest = idx >> 10;          // (b*K+k)*DI + d
  const int d = rest % DI_;
  const int bk = rest / DI_;
  const int kd = bk & 3;
  const int C_ = R_ + 2 * N_;
  float acc = dt_b[kd * DI_ + d];
  const float* wv = dt_w + (size_t)(kd * DI_ + d) * R_;
  const float* yv = ynat + (((size_t)bk * C_) << 10) + p;
#pragma unroll
  for (int r = 0; r < R_; ++r) acc += wv[r] * yv[(size_t)r << 10];
  delta[(size_t)idx] = (acc > 20.0f) ? acc : log1pf(expf(acc));
}

// ---------------------------------------------------------------------------
// Kernel 6: selective scan.  One block per (b,k); thread d holds h[3][16].
// The per-step B/C broadcasts (the latency-critical loads on the serial
// recurrence chain) are staged into LDS with double-buffered
// global_load_async_to_lds_b32 (ASYNCcnt), so tile t+1's DMA overlaps
// tile t's recurrence.  Per-thread delta/u streams stay in global and get
// global_prefetch hints one tile ahead.
// grid = B*K = 8, block = DI = 192.
// ---------------------------------------------------------------------------
__global__ __launch_bounds__(192) void k_scan(const float* __restrict__ delta,
                                              const float* __restrict__ ynat,
                                              const float* __restrict__ A_logs,
                                              const float* __restrict__ Ds,
                                              const float* __restrict__ u0,
                                              const float* __restrict__ u1,
                                              const float* __restrict__ u2,
                                              float* __restrict__ yscan) {
  __shared__ float sBC[2][2 * N_ * TL_];   // [buf][{B,C}*N*TL + n*TL + j] 16KB

  const int bk = blockIdx.x;          // b*K + k
  const int kd = bk & 3;
  const int bi = bk >> 2;
  const int d = threadIdx.x;
  const int C_ = R_ + 2 * N_;
  const int NTILE = L_ / TL_;
  const int EPT = (2 * N_ * TL_ + DI_ - 1) / DI_;  // async elems per thread

  float An[N_];
#pragma unroll
  for (int n = 0; n < N_; ++n)
    An[n] = -expf(A_logs[(size_t)(kd * DI_ + d) * N_ + n]);
  const float Dv = Ds[kd * DI_ + d];

  float h[3][N_];
#pragma unroll
  for (int m = 0; m < 3; ++m)
#pragma unroll
    for (int n = 0; n < N_; ++n) h[m][n] = 0.0f;

  const float* dptr = delta + (((size_t)bk * DI_ + d) << 10);
  const float* Bb = ynat + (((size_t)bk * C_ + R_) << 10);
  const float* Cb = ynat + (((size_t)bk * C_ + R_ + N_) << 10);
  const float* ub[3] = {u0 + (((size_t)(bi * DI_ + d)) << 10),
                        u1 + (((size_t)(bi * DI_ + d)) << 10),
                        u2 + (((size_t)(bi * DI_ + d)) << 10)};
  float* yo[3];
#pragma unroll
  for (int m = 0; m < 3; ++m)
    yo[m] = yscan + ((((size_t)(m * B_ + bi) * K_ + kd) * DI_ + d) << 10);

  // Stage tile t's B/C values (in direction-l order) into LDS buffer `buf`.
  auto stage_tile = [&](int t, int buf) {
    const int l0 = t * TL_;
#pragma unroll
    for (int i = 0; i < EPT; ++i) {
      const int e = d + i * DI_;
      if (e < 2 * N_ * TL_) {
        const int which = e / (N_ * TL_);        // 0 = B, 1 = C
        const int rem = e - which * (N_ * TL_);
        const int n = rem / TL_;
        const int j = rem - n * TL_;
        const int p = perm_k(kd, l0 + j);
        const float* ga = (which ? Cb : Bb) + (((size_t)n) << 10) + p;
        // low 32 bits of a flat LDS pointer == LDS byte address
        const unsigned lds = (unsigned)(unsigned long long)(const void*)&sBC[buf][e];
        async_ld_b32(lds, ga);
      }
    }
    // prefetch hints for next tile's per-thread streams
    const int pf0 = perm_k(kd, l0);
    const int pf1 = perm_k(kd, l0 + TL_ / 2);
    __builtin_prefetch(dptr + pf0, 0, 3);
    __builtin_prefetch(dptr + pf1, 0, 3);
#pragma unroll
    for (int m = 0; m < 3; ++m) {
      __builtin_prefetch(ub[m] + pf0, 0, 3);
      __builtin_prefetch(ub[m] + pf1, 0, 3);
    }
  };

  stage_tile(0, 0);
  for (int t = 0; t < NTILE; ++t) {
    wait_asynccnt0();       // this wave's staged loads complete
    __syncthreads();        // all waves' loads visible; prev readers done
    if (t + 1 < NTILE) stage_tile(t + 1, (t + 1) & 1);

    const float* Bs = &sBC[t & 1][0];
    const float* Cs = &sBC[t & 1][N_ * TL_];
    for (int j = 0; j < TL_; ++j) {
      const int l = t * TL_ + j;
      const int p = perm_k(kd, l);
      const float dv = dptr[p];
      float dA[N_];
#pragma unroll
      for (int n = 0; n < N_; ++n) dA[n] = expf(dv * An[n]);
#pragma unroll
      for (int m = 0; m < 3; ++m) {
        const float uv = ub[m][p];
        const float du = dv * uv;
        float y = 0.0f;
#pragma unroll
        for (int n = 0; n < N_; ++n) {
          h[m][n] = dA[n] * h[m][n] + du * Bs[n * TL_ + j];
          y += h[m][n] * Cs[n * TL_ + j];
        }
        yo[m][l] = y + Dv * uv;
      }
    }
  }
}

// ---------------------------------------------------------------------------
// Kernel 7: direction recombination + LayerNorm(DI) + SiLU(z) gate.
//   g[m,b,p,d] = LN(ysum)[d] * silu(z_m[b,p,d])
// grid = 3*B*L blocks, block = DI = 192 (two-pass shared reduction).
// ---------------------------------------------------------------------------
__global__ __launch_bounds__(192) void k_combine_ln(const float* __restrict__ yscan,
                                                    const float* __restrict__ z0,
                                                    const float* __restrict__ z1,
                                                    const float* __restrict__ z2,
                                                    const float* __restrict__ ln_w,
                                                    const float* __restrict__ ln_b,
                                                    float* __restrict__ g) {
  __shared__ float s[DI_];
  const int mbp = blockIdx.x;
  const int p = mbp & (L_ - 1);
  const int mb = mbp >> 10;
  const int bi = mb % B_;
  const int m = mb / B_;
  const int d = threadIdx.x;
  const int hh = p >> 5, ww = p & 31;
  const int l1 = ww * 32 + hh;          // col-major index of (hh,ww)

  const size_t base = (((size_t)(m * B_ + bi) * K_) * DI_ + d) << 10;
  const size_t kstep = (size_t)DI_ << 10;
  float t = yscan[base + p]                                   // k=0
          + yscan[base + kstep + l1]                          // k=1 (wh)
          + yscan[base + 2 * kstep + (L_ - 1 - p)]            // k=2 (flip)
          + yscan[base + 3 * kstep + (L_ - 1 - l1)];          // k=3 (wh∘flip)

  // mean
  s[d] = t;
  __syncthreads();
  for (int off = 128; off > 0; off >>= 1) {
    if (d < off && d + off < DI_) s[d] += s[d + off];
    __syncthreads();
  }
  const float mean = s[0] * (1.0f / DI_);
  __syncthreads();
  // variance (biased, matches jnp.var)
  const float ct = t - mean;
  s[d] = ct * ct;
  __syncthreads();
  for (int off = 128; off > 0; off >>= 1) {
    if (d < off && d + off < DI_) s[d] += s[d + off];
    __syncthreads();
  }
  const float var = s[0] * (1.0f / DI_);

  const float* zz = (m == 0) ? z0 : ((m == 1) ? z1 : z2);
  const float zr = zz[(((size_t)bi << 10) + p) * DI_ + d];
  const float norm = ct * rsqrtf(var + 1e-5f) * ln_w[d] + ln_b[d];
  g[(((size_t)(m * B_ + bi) << 10) + p) * DI_ + d] = norm * silu_(zr);
}

// ---------------------------------------------------------------------------
// Kernel 8: output projection GEMM per branch.
//   out[m,b,p,c] = sum_d g[m,b,p,d] * out_w_m[c,d]
// grid (128 Mtiles, 6 Ntiles, 3 branches), block 32.
// ---------------------------------------------------------------------------
__global__ __launch_bounds__(32) void k_outproj(const float* __restrict__ g,
                                                const float* __restrict__ ow0,
                                                const float* __restrict__ ow1,
                                                const float* __restrict__ ow2,
                                                float* __restrict__ out) {
  const int lane = threadIdx.x & 31;
  const int hf = lane >> 4, lr = lane & 15;
  const int m0 = blockIdx.x * 16;  // row i = b*L + p
  const int n0 = blockIdx.y * 16;  // col c
  const int t = blockIdx.z;
  const float* gt = g + (size_t)t * (B_ * L_) * DI_;
  const float* ow = (t == 0) ? ow0 : ((t == 1) ? ow1 : ow2);
  v8f c = {};
  const float* arow = gt + (size_t)(m0 + lr) * DI_;
  const float* brow = ow + (size_t)(n0 + lr) * DI_;
#pragma unroll 4
  for (int k0 = 0; k0 < DI_; k0 += 4) {
    const int kb = k0 + 2 * hf;
    v2f a, b;
    a.x = arow[kb];  a.y = arow[kb + 1];
    b.x = brow[kb];  b.y = brow[kb + 1];
    c = wmma4(a, b, c);
  }
#pragma unroll
  for (int v = 0; v < 8; ++v) {
    const int i = m0 + v + 8 * hf;
    out[((size_t)t * (B_ * L_) + i) * DM_ + (n0 + lr)] = c[v];
  }
}

// ---------------------------------------------------------------------------
extern "C" void kernel_launch(void* const* d_in, const int* in_sizes, int n_in,
                              void* d_out, int out_size, void* d_ws, size_t ws_size,
                              hipStream_t stream) {
  const float* x[3]  = {(const float*)d_in[0], (const float*)d_in[1], (const float*)d_in[2]};
  const float* iw[3] = {(const float*)d_in[3], (const float*)d_in[4], (const float*)d_in[5]};
  const float* cw[3] = {(const float*)d_in[6], (const float*)d_in[8], (const float*)d_in[10]};
  const float* cb[3] = {(const float*)d_in[7], (const float*)d_in[9], (const float*)d_in[11]};
  const float* fuse_w = (const float*)d_in[12];
  const float* xpw    = (const float*)d_in[13];
  const float* dt_w   = (const float*)d_in[14];
  const float* dt_b   = (const float*)d_in[15];
  const float* A_logs = (const float*)d_in[16];
  const float* Ds     = (const float*)d_in[17];
  const float* ln_w   = (const float*)d_in[18];
  const float* ln_b   = (const float*)d_in[19];
  const float* ow[3]  = {(const float*)d_in[20], (const float*)d_in[21], (const float*)d_in[22]};
  float* out = (float*)d_out;

  // Workspace carve-up (floats). Total ~11.7M floats (~47 MB).
  const size_t BDL = (size_t)B_ * DI_ * L_;                // 393216
  const size_t YN  = (size_t)B_ * K_ * (R_ + 2 * N_) * L_; // 311296
  const size_t DLT = (size_t)B_ * K_ * DI_ * L_;           // 1572864
  float* ws = (float*)d_ws;
  float* xc_raw = ws;                 ws += 3 * BDL;
  float* xc_act = ws;                 ws += 3 * BDL;
  float* zb     = ws;                 ws += 3 * BDL;
  float* xfuse  = ws;                 ws += BDL;
  float* ynat   = ws;                 ws += YN;
  float* delta  = ws;                 ws += DLT;
  float* yscan  = ws;                 ws += 3 * DLT;
  float* gbuf   = ws;                 ws += 3 * BDL;

  // 1) input projections (WMMA f32) + split, per branch
  for (int t = 0; t < 3; ++t)
    k_inproj<<<dim3((B_ * L_) / 16, (2 * DI_) / 16), 32, 0, stream>>>(
        x[t], iw[t], xc_raw + (size_t)t * BDL, zb + (size_t)t * BDL);

  // 2) depthwise 3x3 conv + SiLU, per branch
  for (int t = 0; t < 3; ++t)
    k_conv_silu<<<dim3((unsigned)(BDL / 256)), 256, 0, stream>>>(
        xc_raw + (size_t)t * BDL, cw[t], cb[t], xc_act + (size_t)t * BDL);

  // 3) cross-modal fuse GEMM (WMMA f32)
  k_fuse<<<dim3(DI_ / 16, L_ / 16, B_), 32, 0, stream>>>(
      fuse_w, xc_act, xc_act + BDL, xc_act + 2 * BDL, xfuse);

  // 4) dt/B/C projection GEMM in natural order (WMMA f32)
  k_xdbl<<<dim3(3, L_ / 16, B_ * K_), 32, 0, stream>>>(xpw, xfuse, ynat);

  // 5) dt projection + softplus
  k_delta<<<dim3((unsigned)(DLT / 256)), 256, 0, stream>>>(ynat, dt_w, dt_b, delta);

  // 6) selective scan with double-buffered async LDS staging of B/C
  k_scan<<<dim3(B_ * K_), DI_, 0, stream>>>(delta, ynat, A_logs, Ds,
                                            xc_act, xc_act + BDL, xc_act + 2 * BDL,
                                            yscan);

  // 7) direction recombination + LN + SiLU(z) gate
  k_combine_ln<<<dim3(3 * B_ * L_), DI_, 0, stream>>>(
      yscan, zb, zb + BDL, zb + 2 * BDL, ln_w, ln_b, gbuf);

  // 8) output projections (WMMA f32)
  k_outproj<<<dim3((B_ * L_) / 16, DM_ / 16, 3), 32, 0, stream>>>(
      gbuf, ow[0], ow[1], ow[2], out);
}